// GCNN_90022514524343
// MI455X (gfx1250) — compile-verified
//
#include <hip/hip_runtime.h>

#define N_NODES 50000
#define N_EDGES 800000
#define HDIM    64

typedef float v2f __attribute__((ext_vector_type(2)));
typedef float v8f __attribute__((ext_vector_type(8)));

// ---------------- degree / normalization ----------------
__global__ void deg_init(float* __restrict__ deg) {
    int i = blockIdx.x * 256 + threadIdx.x;
    if (i < N_NODES) deg[i] = 1.0f;               // self-loop contributes 1
}

__global__ void deg_count(const int* __restrict__ ei, float* __restrict__ deg) {
    int e = blockIdx.x * 256 + threadIdx.x;
    if (e < N_EDGES) {
        int dst = ei[N_EDGES + e];                // edge_index[1][e]
        atomicAdd(&deg[dst], 1.0f);
    }
}

__global__ void deg_to_dinv(float* __restrict__ deg) {
    int i = blockIdx.x * 256 + threadIdx.x;
    if (i < N_NODES) deg[i] = rsqrtf(deg[i]);     // deg >= 1 always (self-loops)
}

// ---------------- fused GEMM: h = act(in)+b? @ W ; agg = h * dinv^2 ----------------
// One wave computes one 16x16 tile of h using V_WMMA_F32_16X16X4_F32 (K=4, 16 steps).
template <bool RELU_IN>
__global__ __launch_bounds__(128)
void gcn_gemm(const float* __restrict__ in, const float* __restrict__ bias,
              const float* __restrict__ W, const float* __restrict__ dinv,
              float* __restrict__ h, float* __restrict__ agg) {
    const int wave = threadIdx.x >> 5;
    const int lane = threadIdx.x & 31;
    const int tile = blockIdx.x * 4 + wave;       // 0..12499, exact
    const int tm   = tile >> 2;                   // row tile 0..3124
    const int tn   = tile & 3;                    // col tile 0..3

    const int mrow  = (tm << 4) + (lane & 15);    // A row for this lane
    const int khalf = (lane >> 4) << 1;           // 0 or 2 (K sub-slot)
    const int ncol  = (tn << 4) + (lane & 15);    // B col for this lane

    const float* __restrict__ arow = in + (size_t)mrow * HDIM;

    v8f c = {0.f, 0.f, 0.f, 0.f, 0.f, 0.f, 0.f, 0.f};
#pragma unroll
    for (int k = 0; k < HDIM; k += 4) {
        const int ka = k + khalf;
        v2f a;
        a.x = arow[ka];
        a.y = arow[ka + 1];
        if (RELU_IN) {                            // fused prev-layer bias + ReLU
            a.x = fmaxf(a.x + bias[ka],     0.f);
            a.y = fmaxf(a.y + bias[ka + 1], 0.f);
        }
        v2f b;
        b.x = W[(size_t)ka * HDIM + ncol];
        b.y = W[(size_t)(ka + 1) * HDIM + ncol];
        // D = A(16x4) * B(4x16) + C  -- f32, exact
        c = __builtin_amdgcn_wmma_f32_16x16x4_f32(false, a, false, b,
                                                  (short)0, c, false, false);
    }

    // C/D layout: VGPR v -> row (v + 8*(lane>=16)), col = lane&15
    const int mbase = (tm << 4) + ((lane >> 4) << 3);
    const int nn    = (tn << 4) + (lane & 15);
#pragma unroll
    for (int v = 0; v < 8; ++v) {
        const int row = mbase + v;
        const float val = c[v];
        h[(size_t)row * HDIM + nn] = val;
        const float di = dinv[row];
        agg[(size_t)row * HDIM + nn] = val * di * di;   // fused self-loop term
    }
}

// ---------------- edge scatter: agg[dst] += h[src] * dinv[src]*dinv[dst] ----------------
// 16 threads per edge, float4 per thread (covers all 64 features).
__global__ __launch_bounds__(256)
void gcn_scatter(const int* __restrict__ ei, const float* __restrict__ dinv,
                 const float* __restrict__ h, float* __restrict__ agg) {
    const int t = blockIdx.x * 256 + threadIdx.x;
    const int e = t >> 4;
    if (e >= N_EDGES) return;
    const int fo  = (t & 15) << 2;
    const int src = ei[e];
    const int dst = ei[N_EDGES + e];
    const float nrm = dinv[src] * dinv[dst];

    const float4 v = *(const float4*)(h + (size_t)src * HDIM + fo);
    float* ap = agg + (size_t)dst * HDIM + fo;
    atomicAdd(ap + 0, v.x * nrm);
    atomicAdd(ap + 1, v.y * nrm);
    atomicAdd(ap + 2, v.z * nrm);
    atomicAdd(ap + 3, v.w * nrm);
}

// ---------------- final activation ----------------
__global__ void gcn_act(const float* __restrict__ agg, const float* __restrict__ bias,
                        float* __restrict__ out) {
    int i = blockIdx.x * 256 + threadIdx.x;
    if (i < N_NODES * HDIM)
        out[i] = fmaxf(agg[i] + bias[i & (HDIM - 1)], 0.f);
}

extern "C" void kernel_launch(void* const* d_in, const int* in_sizes, int n_in,
                              void* d_out, int out_size, void* d_ws, size_t ws_size,
                              hipStream_t stream) {
    const float* x  = (const float*)d_in[0];   // [N,H]
    const int*   ei = (const int*)d_in[1];     // [2,E]
    const float* W  = (const float*)d_in[2];   // [L,H,H]
    const float* b  = (const float*)d_in[3];   // [L,H]
    float* out = (float*)d_out;

    float* ws   = (float*)d_ws;
    float* dinv = ws;                                   // N
    float* h    = dinv + N_NODES;                       // N*H
    float* aggA = h    + (size_t)N_NODES * HDIM;        // N*H
    float* aggB = aggA + (size_t)N_NODES * HDIM;        // N*H

    const int nodeBlocks = (N_NODES + 255) / 256;
    const int edgeBlocks = (N_EDGES + 255) / 256;
    const int scatBlocks = (N_EDGES * 16) / 256;        // exact: 50000
    const int gemmBlocks = (N_NODES / 16) * (HDIM / 16) / 4;  // exact: 3125
    const int actBlocks  = (N_NODES * HDIM + 255) / 256;

    // normalization: deg -> dinv (in place)
    deg_init   <<<nodeBlocks, 256, 0, stream>>>(dinv);
    deg_count  <<<edgeBlocks, 256, 0, stream>>>(ei, dinv);
    deg_to_dinv<<<nodeBlocks, 256, 0, stream>>>(dinv);

    // layer 0: x -> h -> aggA
    gcn_gemm<false><<<gemmBlocks, 128, 0, stream>>>(x, nullptr, W, dinv, h, aggA);
    gcn_scatter    <<<scatBlocks, 256, 0, stream>>>(ei, dinv, h, aggA);

    // layer 1: relu(aggA+b0) -> h -> aggB
    gcn_gemm<true> <<<gemmBlocks, 128, 0, stream>>>(aggA, b, W + HDIM * HDIM, dinv, h, aggB);
    gcn_scatter    <<<scatBlocks, 256, 0, stream>>>(ei, dinv, h, aggB);

    // layer 2: relu(aggB+b1) -> h -> aggA
    gcn_gemm<true> <<<gemmBlocks, 128, 0, stream>>>(aggB, b + HDIM, W + 2 * HDIM * HDIM, dinv, h, aggA);
    gcn_scatter    <<<scatBlocks, 256, 0, stream>>>(ei, dinv, h, aggA);

    // final: out = relu(aggA + b2)
    gcn_act<<<actBlocks, 256, 0, stream>>>(aggA, b + 2 * HDIM, out);
}